// CustomActivation_1546188226841
// MI455X (gfx1250) — compile-verified
//
#include <hip/hip_runtime.h>

// SiLU: y = x * sigmoid(x), f32 -> f32, 4x4096x4096 = 67,108,864 elements.
// Memory-bound streaming kernel (537 MB traffic, ~23 us floor @ 23.3 TB/s).
// - 128-bit non-temporal loads/stores (working set > 192 MB L2, single touch)
// - 4 independent b128 transactions per thread for memory-level parallelism
// - wave-uniform full-block test so the bulk path has NO exec-mask predication
// - wave32-friendly 256-thread blocks (8 waves/block)

typedef float v4f __attribute__((ext_vector_type(4)));

#ifndef SILU_VECS_PER_THREAD
#define SILU_VECS_PER_THREAD 4   // 4 x float4 = 64 B per thread
#endif

__device__ __forceinline__ float silu_scalar(float x) {
    // sigmoid(x) = 1 / (1 + exp(-x)); __expf -> v_exp_f32 (TRANS32, co-executes
    // with VALU), __builtin_amdgcn_rcpf -> v_rcp_f32. ~1 ulp each.
    float e = __expf(-x);
    float s = __builtin_amdgcn_rcpf(1.0f + e);
    return x * s;
}

__device__ __forceinline__ v4f silu_vec4(v4f v) {
    v4f r;
    r.x = silu_scalar(v.x);
    r.y = silu_scalar(v.y);
    r.z = silu_scalar(v.z);
    r.w = silu_scalar(v.w);
    return r;
}

__global__ __launch_bounds__(256) void silu_f32_kernel(const v4f* __restrict__ in,
                                                       v4f* __restrict__ out,
                                                       unsigned int nvec) {
    const unsigned int vpb       = 256u * SILU_VECS_PER_THREAD;
    const unsigned int blockBase = blockIdx.x * vpb;       // scalar
    const unsigned int base      = blockBase + threadIdx.x;

    if (blockBase + vpb <= nvec) {
        // Fast path (all blocks except possibly the last): scalar branch,
        // no EXEC predication. Issue all 4 NT b128 loads back-to-back.
        v4f val[SILU_VECS_PER_THREAD];
#pragma unroll
        for (int k = 0; k < SILU_VECS_PER_THREAD; ++k) {
            val[k] = __builtin_nontemporal_load(&in[base + (unsigned)k * 256u]);
        }
#pragma unroll
        for (int k = 0; k < SILU_VECS_PER_THREAD; ++k) {
            v4f r = silu_vec4(val[k]);
            __builtin_nontemporal_store(r, &out[base + (unsigned)k * 256u]);
        }
    } else {
        // Boundary block only: per-vector guard.
#pragma unroll
        for (int k = 0; k < SILU_VECS_PER_THREAD; ++k) {
            unsigned int i = base + (unsigned)k * 256u;
            if (i < nvec) {
                v4f v = __builtin_nontemporal_load(&in[i]);
                __builtin_nontemporal_store(silu_vec4(v), &out[i]);
            }
        }
    }
}

// Scalar tail kernel for element counts not divisible by 4 (not hit for
// 4*4096*4096, but keeps the launcher correct for any size).
__global__ __launch_bounds__(256) void silu_f32_tail_kernel(const float* __restrict__ in,
                                                            float* __restrict__ out,
                                                            unsigned int start,
                                                            unsigned int n) {
    unsigned int i = start + blockIdx.x * 256u + threadIdx.x;
    if (i < n) {
        out[i] = silu_scalar(in[i]);
    }
}

extern "C" void kernel_launch(void* const* d_in, const int* in_sizes, int n_in,
                              void* d_out, int out_size, void* d_ws, size_t ws_size,
                              hipStream_t stream) {
    (void)n_in; (void)d_ws; (void)ws_size; (void)out_size;

    const float* x = (const float*)d_in[0];
    float*       y = (float*)d_out;
    const unsigned int n = (unsigned int)in_sizes[0];

    const unsigned int nvec = n / 4u;                 // full float4s
    const unsigned int vecs_per_block = 256u * SILU_VECS_PER_THREAD;
    if (nvec > 0) {
        unsigned int grid = (nvec + vecs_per_block - 1u) / vecs_per_block;
        silu_f32_kernel<<<grid, 256, 0, stream>>>((const v4f*)x, (v4f*)y, nvec);
    }

    const unsigned int tail_start = nvec * 4u;
    if (tail_start < n) {
        unsigned int tail = n - tail_start;
        unsigned int grid = (tail + 255u) / 256u;
        silu_f32_tail_kernel<<<grid, 256, 0, stream>>>(x, y, tail_start, n);
    }
}